// ResidualGraphConv_58634893525278
// MI455X (gfx1250) — compile-verified
//
#include <hip/hip_runtime.h>
#include <hip/hip_bf16.h>

typedef __attribute__((ext_vector_type(2))) float v2f;
typedef __attribute__((ext_vector_type(8))) float v8f;

#define D 128
#define EPS 1e-5f

// ---------------- LayerNorm: one wave (32 lanes) per row, float4 per lane ----
__global__ __launch_bounds__(256) void ln_kernel(const float* __restrict__ x,
                                                 const float* __restrict__ w,
                                                 const float* __restrict__ b,
                                                 float* __restrict__ h, int n) {
    int wave = threadIdx.x >> 5;
    int lane = threadIdx.x & 31;
    int row = blockIdx.x * 8 + wave;
    if (row >= n) return;
    const float4* xr = (const float4*)(x + (size_t)row * D);
    float4 v = xr[lane];
    float s = v.x + v.y + v.z + v.w;
    #pragma unroll
    for (int o = 16; o > 0; o >>= 1) s += __shfl_xor(s, o, 32);
    float mu = s * (1.0f / D);
    float d0 = v.x - mu, d1 = v.y - mu, d2 = v.z - mu, d3 = v.w - mu;
    float q = d0 * d0 + d1 * d1 + d2 * d2 + d3 * d3;
    #pragma unroll
    for (int o = 16; o > 0; o >>= 1) q += __shfl_xor(q, o, 32);
    float rstd = __frsqrt_rn(q * (1.0f / D) + EPS);
    float4 wv = ((const float4*)w)[lane];
    float4 bv = ((const float4*)b)[lane];
    float4 o4;
    o4.x = d0 * rstd * wv.x + bv.x;
    o4.y = d1 * rstd * wv.y + bv.y;
    o4.z = d2 * rstd * wv.z + bv.z;
    o4.w = d3 * rstd * wv.w + bv.w;
    ((float4*)(h + (size_t)row * D))[lane] = o4;
}

// ---------------- in-degree (excluding self loop) ---------------------------
__global__ __launch_bounds__(256) void deg_kernel(const long long* __restrict__ col,
                                                  float* __restrict__ deg, int E) {
    int e = blockIdx.x * blockDim.x + threadIdx.x;
    if (e < E) atomicAdd(&deg[col[e]], 1.0f);
}

// ---------------- dinv = rsqrt(deg + 1)  (self loop folded in) --------------
__global__ __launch_bounds__(256) void dinv_kernel(float* __restrict__ deg, int n) {
    int i = blockIdx.x * blockDim.x + threadIdx.x;
    if (i < n) deg[i] = __frsqrt_rn(deg[i] + 1.0f);
}

// ---------------- hw = h @ W with fp32 WMMA ---------------------------------
// 256 threads = 8 waves; block covers 16 rows; wave w owns cols [16w, 16w+16).
// K=128 -> 32 chained v_wmma_f32_16x16x4_f32 per wave.
__global__ __launch_bounds__(256) void gemm_wmma_kernel(const float* __restrict__ h,
                                                        const float* __restrict__ W,
                                                        float* __restrict__ hw) {
    int wave = threadIdx.x >> 5;
    int lane = threadIdx.x & 31;
    int m = lane & 15;        // A row / B,C,D col within tile
    int kh = lane >> 4;       // K half (0 or 1)
    int row0 = blockIdx.x * 16;
    int col0 = wave * 16;

    const float* __restrict__ arow = h + (size_t)(row0 + m) * D;
    v8f acc = {};
    #pragma unroll 4
    for (int k = 0; k < D; k += 4) {
        int kk = k + 2 * kh;
        v2f a, b;
        a.x = arow[kk + 0];
        a.y = arow[kk + 1];
        b.x = W[(size_t)(kk + 0) * D + col0 + m];
        b.y = W[(size_t)(kk + 1) * D + col0 + m];
        acc = __builtin_amdgcn_wmma_f32_16x16x4_f32(
            /*neg_a=*/false, a, /*neg_b=*/false, b,
            /*c_mod=*/(short)0, acc, /*reuse_a=*/false, /*reuse_b=*/false);
    }
    // C/D layout: VGPR v -> M = v + 8*kh, N = lane&15
    #pragma unroll
    for (int v = 0; v < 8; ++v) {
        int r = row0 + v + 8 * kh;
        hw[(size_t)r * D + col0 + m] = acc[v];
    }
}

// ---------------- edge scatter: one wave per edge, float4 of atomics/lane ---
__global__ __launch_bounds__(256) void scatter_kernel(const long long* __restrict__ ei,
                                                      const float* __restrict__ dinv,
                                                      const float* __restrict__ hw,
                                                      float* __restrict__ accum, int E) {
    int ew = (int)((blockIdx.x * (size_t)blockDim.x + threadIdx.x) >> 5);
    int lane = threadIdx.x & 31;
    if (ew >= E) return;
    long long r = ei[ew];
    long long c = ei[(size_t)E + ew];
    float norm = dinv[r] * dinv[c];
    float4 mv = ((const float4*)(hw + (size_t)r * D))[lane];
    float* dst = accum + (size_t)c * D + lane * 4;
    atomicAdd(dst + 0, mv.x * norm);
    atomicAdd(dst + 1, mv.y * norm);
    atomicAdd(dst + 2, mv.z * norm);
    atomicAdd(dst + 3, mv.w * norm);
}

// ---------------- final: out = relu(x + accum + hw*dinv^2 + b) --------------
// hw lives in d_out; each thread reads its own element then overwrites it.
__global__ __launch_bounds__(256) void final_kernel(const float* __restrict__ x,
                                                    const float* __restrict__ accum,
                                                    const float* __restrict__ dinv,
                                                    const float* __restrict__ bconv,
                                                    float* __restrict__ out, int n) {
    size_t i = blockIdx.x * (size_t)blockDim.x + threadIdx.x;
    if (i >= (size_t)n * D) return;
    int node = (int)(i >> 7);
    int c = (int)(i & (D - 1));
    float di = dinv[node];
    float v = x[i] + accum[i] + out[i] * di * di + bconv[c];
    out[i] = v > 0.0f ? v : 0.0f;
}

extern "C" void kernel_launch(void* const* d_in, const int* in_sizes, int n_in,
                              void* d_out, int out_size, void* d_ws, size_t ws_size,
                              hipStream_t stream) {
    const float*     x    = (const float*)d_in[0];
    const long long* ei   = (const long long*)d_in[1];   // int64 [2, E]
    const float*     lnw  = (const float*)d_in[2];
    const float*     lnb  = (const float*)d_in[3];
    const float*     W    = (const float*)d_in[4];
    const float*     bcv  = (const float*)d_in[5];
    float*           out  = (float*)d_out;

    const int N = in_sizes[0] / D;       // 50000
    const int E = in_sizes[1] / 2;       // 640000

    // workspace layout
    char* ws = (char*)d_ws;
    float* h     = (float*)ws;                                   // N*D floats
    float* accum = (float*)(ws + (size_t)N * D * sizeof(float)); // N*D floats
    float* deg   = accum + (size_t)N * D;                        // N floats (contiguous after accum)

    // zero accum + deg each call (graph-capture-safe memset node)
    hipMemsetAsync(accum, 0, ((size_t)N * D + N) * sizeof(float), stream);

    // 1) layernorm
    ln_kernel<<<(N + 7) / 8, 256, 0, stream>>>(x, lnw, lnb, h, N);
    // 2) degrees from targets (col = second row of edge_index)
    deg_kernel<<<(E + 255) / 256, 256, 0, stream>>>(ei + E, deg, E);
    // 3) dinv
    dinv_kernel<<<(N + 255) / 256, 256, 0, stream>>>(deg, N);
    // 4) hw = h @ W  (fp32 WMMA), staged in d_out
    gemm_wmma_kernel<<<N / 16, 256, 0, stream>>>(h, W, out);
    // 5) edge scatter-add into accum
    scatter_kernel<<<(E * 32 + 255) / 256, 256, 0, stream>>>(ei, deg, out, accum, E);
    // 6) fused self-loop + bias + residual + relu (reads d_out as hw, overwrites)
    final_kernel<<<((size_t)N * D + 255) / 256, 256, 0, stream>>>(x, accum, deg, bcv, out, N);
}